// TargetSimplex_66030827209360
// MI455X (gfx1250) — compile-verified
//
#include <hip/hip_runtime.h>
#include <hip/hip_bf16.h>
#include <math.h>
#include <stdint.h>

// Problem sizes (fixed by the reference).
#define B_   4
#define S_   128
#define V_   50257
#define D_   768
#define ROWS (B_ * S_)   // 512

typedef __attribute__((ext_vector_type(2))) float        v2f;
typedef __attribute__((ext_vector_type(8))) float        v8f;
typedef __attribute__((ext_vector_type(4))) unsigned int v4u;
typedef __attribute__((ext_vector_type(4))) int          v4i;
typedef __attribute__((ext_vector_type(8))) int          v8i;

#if defined(__has_builtin)
# if __has_builtin(__builtin_amdgcn_tensor_load_to_lds)
#  define HAVE_TDM 1
# endif
#endif

// ---------------------------------------------------------------------------
// Kernel 1: per-row fused argmax + softmax + straight-through pred_probs.
// One 256-thread block per (b,s) row of V=50257 logits.
// ---------------------------------------------------------------------------
__global__ __launch_bounds__(256) void softmax_argmax_kernel(
    const float* __restrict__ logits,      // [ROWS, V]
    float*       __restrict__ out_softmax, // [ROWS, V]
    float*       __restrict__ out_pred,    // [ROWS, V]
    float*       __restrict__ out_index,   // [ROWS]
    int*         __restrict__ ws_idx,      // [ROWS]
    float*       __restrict__ ws_c)        // [ROWS]
{
    const int row = blockIdx.x;
    const int t   = threadIdx.x;
    const float* x = logits + (size_t)row * V_;
    const float4* x4 = (const float4*)x;
    const int V4 = V_ / 4;               // 12564 float4's + 1 tail element

    __shared__ float sred[256];
    __shared__ int   sidx[256];
    __shared__ float s_bmax;
    __shared__ int   s_bidx;
    __shared__ float s_binv;

    // ---- pass 1: max / argmax (first-occurrence tie-break) ----
    float m = -INFINITY;
    int   mi = 0;
    for (int i = t; i < V4; i += 256) {
        float4 v = x4[i];
        const int base = i * 4;
        if (v.x > m) { m = v.x; mi = base;     }
        if (v.y > m) { m = v.y; mi = base + 1; }
        if (v.z > m) { m = v.z; mi = base + 2; }
        if (v.w > m) { m = v.w; mi = base + 3; }
    }
    for (int i = V4 * 4 + t; i < V_; i += 256) {
        float v = x[i];
        if (v > m) { m = v; mi = i; }
    }
    sred[t] = m; sidx[t] = mi;
    __syncthreads();
    for (int off = 128; off > 0; off >>= 1) {
        if (t < off) {
            float om = sred[t + off]; int oi = sidx[t + off];
            if (om > sred[t] || (om == sred[t] && oi < sidx[t])) {
                sred[t] = om; sidx[t] = oi;
            }
        }
        __syncthreads();
    }
    if (t == 0) { s_bmax = sred[0]; s_bidx = sidx[0]; }
    __syncthreads();
    const float bmax = s_bmax;
    const int   bidx = s_bidx;
    __syncthreads();   // sred reused below

    // ---- pass 2: sum of exp(x - max) ----
    float sum = 0.0f;
    for (int i = t; i < V4; i += 256) {
        float4 v = x4[i];
        sum += __expf(v.x - bmax) + __expf(v.y - bmax)
             + __expf(v.z - bmax) + __expf(v.w - bmax);
    }
    for (int i = V4 * 4 + t; i < V_; i += 256)
        sum += __expf(x[i] - bmax);
    sred[t] = sum;
    __syncthreads();
    for (int off = 128; off > 0; off >>= 1) {
        if (t < off) sred[t] += sred[t + off];
        __syncthreads();
    }
    if (t == 0) {
        const float inv = 1.0f / sred[0];
        s_binv = inv;
        const float p = inv;                   // softmax at argmax = exp(0)/sum
        const float c = (1.0f - p) + p;        // straight-through forward value
        ws_idx[row]  = bidx;
        ws_c[row]    = c;
        out_index[row] = (float)bidx;
    }
    __syncthreads();
    const float inv  = s_binv;
    const float cval = (1.0f - inv) + inv;

    // ---- pass 3: write softmax and pred_probs ----
    float4* o4 = (float4*)(out_softmax + (size_t)row * V_);
    float4* p4 = (float4*)(out_pred    + (size_t)row * V_);
    for (int i = t; i < V4; i += 256) {
        float4 v = x4[i];
        float4 sm;
        sm.x = __expf(v.x - bmax) * inv;
        sm.y = __expf(v.y - bmax) * inv;
        sm.z = __expf(v.z - bmax) * inv;
        sm.w = __expf(v.w - bmax) * inv;
        o4[i] = sm;
        float4 pp = make_float4(0.f, 0.f, 0.f, 0.f);
        const int base = i * 4;
        const int d = bidx - base;
        if (d >= 0 && d < 4) ((float*)&pp)[d] = cval;
        p4[i] = pp;
    }
    for (int i = V4 * 4 + t; i < V_; i += 256) {
        out_softmax[(size_t)row * V_ + i] = __expf(x[i] - bmax) * inv;
        out_pred[(size_t)row * V_ + i]    = (i == bidx) ? cval : 0.0f;
    }
}

// ---------------------------------------------------------------------------
// Kernel 2: compacted einsum via fp32 WMMA + TDM gather.
// pred_probs is a scaled one-hot, so per 16-row M-tile the dense [16,V]x[V,16]
// product reduces exactly to diag(c) x gathered_embed_rows[16,16], expressed
// with 4x V_WMMA_F32_16X16X4_F32. The 16-row embed gather goes through the
// Tensor Data Mover in gather mode (32-bit indices, 8 rows/descriptor,
// 2 descriptors per tile) into LDS; synchronized with s_wait_tensorcnt.
// After the wait, btile's address is escaped into an opaque asm with a
// "memory" clobber: the compiler must assume the DMA wrote it (keeps the
// ds_loads real and ordered) while the accesses stay plain AS(3) ds_load_b32.
// One wave32 block per (N-tile, M-tile, which-embed).
// ---------------------------------------------------------------------------
__global__ __launch_bounds__(32) void onehot_wmma_gemm_kernel(
    const float* __restrict__ embed_content, // [V, D]
    const float* __restrict__ embed_style,   // [V, D]
    const int*   __restrict__ ws_idx,        // [ROWS]
    const float* __restrict__ ws_c,          // [ROWS]
    float*       __restrict__ out_src,       // [ROWS, D]
    float*       __restrict__ out_tgt)       // [ROWS, D]
{
    __shared__ float btile[16 * 16];  // gathered B' tile; sole LDS object -> offset 0

    const int lane  = threadIdx.x;
    const int ntile = blockIdx.x;     // 0..47  (D/16)
    const int mtile = blockIdx.y;     // 0..31  (ROWS/16)
    const int which = blockIdx.z;     // 0: content, 1: style

    const float* emb = which ? embed_style : embed_content;
    float*       out = which ? out_tgt     : out_src;

    const int n0    = ntile * 16;
    const int rbase = mtile * 16;
    const int half  = lane >> 4;      // 0: lanes 0-15, 1: lanes 16-31
    const int lm    = lane & 15;

    // Per-lane straight-through scale (row lm of this M-tile), coalesced load.
    const float cself = ws_c[rbase + lm];

#if HAVE_TDM
    // Uniform (scalar) index loads -> descriptor operands live in SGPRs.
    int idx[16];
#pragma unroll
    for (int k = 0; k < 16; ++k) idx[k] = ws_idx[rbase + k];

    const unsigned long long ga =
        (unsigned long long)(uintptr_t)emb + (unsigned long long)n0 * 4ull;

    // D# group 0: count=1, gather_index_size=32b (bit30), gather_mode=1 (bit31),
    // lds_addr, global_addr[56:0], type=2 at bits 127:126.
    v4u g0a = { 0x1u | (1u << 30) | (1u << 31),
                0u,
                (unsigned)(ga & 0xFFFFFFFFu),
                (unsigned)((ga >> 32) & 0x1FFFFFFull) | (2u << 30) };
    v4u g0b = g0a;
    g0b.y   = 8u * 16u * 4u;   // second half of btile: +512 bytes

    // D# group 1: data_size=4B; tensor_dim0=16; tensor_dim1=V; tile_dim0=16;
    // tile_dim1=8 valid indices; tensor_dim0_stride=D.
    v8i g1 = { (int)0x00020000,
               (int)(16u << 16),
               (int)(((unsigned)V_ & 0xFFFFu) << 16),
               (int)(16u << 16),
               8,
               D_,
               0,
               0 };

    v4i g2a = { idx[0],  idx[1],  idx[2],  idx[3]  };
    v4i g3a = { idx[4],  idx[5],  idx[6],  idx[7]  };
    v4i g2b = { idx[8],  idx[9],  idx[10], idx[11] };
    v4i g3b = { idx[12], idx[13], idx[14], idx[15] };

# if defined(__clang_major__) && __clang_major__ >= 23
    v8i gz = {};
    __builtin_amdgcn_tensor_load_to_lds(g0a, g1, g2a, g3a, gz, 0);
    __builtin_amdgcn_tensor_load_to_lds(g0b, g1, g2b, g3b, gz, 0);
# else
    __builtin_amdgcn_tensor_load_to_lds(g0a, g1, g2a, g3a, 0);
    __builtin_amdgcn_tensor_load_to_lds(g0b, g1, g2b, g3b, 0);
# endif
    __builtin_amdgcn_s_wait_tensorcnt(0);
    // Escape btile into an opaque asm + memory clobber: the compiler must
    // assume the TDM wrote it; reads below stay plain ds_loads after the wait.
    asm volatile("" :: "v"(btile) : "memory");
#else
    // Fallback: cooperative float4 gather into LDS (2 lanes per row).
    {
        const int r    = lane >> 1;
        const int cb   = (lane & 1) * 8;
        const int rowi = ws_idx[rbase + r];
        const float4* src = (const float4*)(emb + (size_t)rowi * D_ + n0 + cb);
        float4 q0 = src[0];
        float4 q1 = src[1];
        *(float4*)&btile[r * 16 + cb]     = q0;
        *(float4*)&btile[r * 16 + cb + 4] = q1;
    }
#endif
    __syncthreads();

    // WMMA fragments per ISA 7.12.2:
    //   A 16x4 f32 : lane row M = lm; v0 holds K=kk+2*half, v1 holds K+1
    //   B 4x16 f32 : lane col N = lm; same K mapping
    v8f acc = {};
#pragma unroll
    for (int kk = 0; kk < 16; kk += 4) {
        const int k0 = kk + half * 2;
        const int k1 = k0 + 1;

        v2f a;
        a.x = (k0 == lm) ? cself : 0.0f;   // diag(c) -- plain v_cndmask
        a.y = (k1 == lm) ? cself : 0.0f;

        v2f b;
        b.x = btile[k0 * 16 + lm];
        b.y = btile[k1 * 16 + lm];

        acc = __builtin_amdgcn_wmma_f32_16x16x4_f32(
            /*neg_a=*/false, a, /*neg_b=*/false, b,
            /*c_mod=*/(short)0, acc, /*reuse_a=*/false, /*reuse_b=*/false);
    }

    // Store D tile (vr: M=r lanes 0-15, M=r+8 lanes 16-31; N=lm).
    const int mbase = rbase + half * 8;
#pragma unroll
    for (int r = 0; r < 8; ++r) {
        out[(size_t)(mbase + r) * D_ + n0 + lm] = acc[r];
    }
}

// ---------------------------------------------------------------------------
// Launch. d_out layout (flat, return order):
//   [0]                      source_pred_emb  ROWS*D
//   [ROWS*D]                 target_pred_emb  ROWS*D
//   [2*ROWS*D]               softmax_probs    ROWS*V
//   [2*ROWS*D + ROWS*V]      index            ROWS
//   [.. + ROWS]              pred_probs       ROWS*V
// ---------------------------------------------------------------------------
extern "C" void kernel_launch(void* const* d_in, const int* in_sizes, int n_in,
                              void* d_out, int out_size, void* d_ws, size_t ws_size,
                              hipStream_t stream) {
    const float* logits  = (const float*)d_in[0];  // [B,S,V]
    const float* content = (const float*)d_in[1];  // [V,D]
    const float* style   = (const float*)d_in[2];  // [V,D]

    float* out = (float*)d_out;
    float* out_src   = out;
    float* out_tgt   = out + (size_t)ROWS * D_;
    float* out_soft  = out + (size_t)2 * ROWS * D_;
    float* out_index = out_soft + (size_t)ROWS * V_;
    float* out_pred  = out_index + ROWS;

    int*   ws_idx = (int*)d_ws;
    float* ws_c   = (float*)((char*)d_ws + ROWS * sizeof(int));

    softmax_argmax_kernel<<<ROWS, 256, 0, stream>>>(
        logits, out_soft, out_pred, out_index, ws_idx, ws_c);

    dim3 grid(D_ / 16, ROWS / 16, 2);
    onehot_wmma_gemm_kernel<<<grid, 32, 0, stream>>>(
        content, style, ws_idx, ws_c, out_src, out_tgt);
}